// QuantumNeuralNetwork_50405736186429
// MI455X (gfx1250) — compile-verified
//
#include <hip/hip_runtime.h>
#include <hip/hip_bf16.h>

// ---------------------------------------------------------------------------
// QuantumNeuralNetwork collapsed to a single affine map:
//   out = x @ W_eff + b_eff
// W_eff = in_W @ E_blockdiag @ M_quad @ meas @ out_W is precomputed on device
// (three small WMMA GEMMs + three tiny kernels, plus one for b_eff), then one 16384x512x512 fp32
// GEMM using V_WMMA_F32_16X16X4_F32 (full-precision matrix pipe on CDNA5).
// Strides are template constants so all inner-loop loads use immediate
// offsets (no per-iteration 64-bit address VALU competing with WMMA issue).
// ---------------------------------------------------------------------------

typedef float v2f __attribute__((ext_vector_type(2)));
typedef float v8f __attribute__((ext_vector_type(8)));

#define HD   512
#define NROW (8 * 2048)

// ---- sup_vec[h] = sum_s coeffs[s] * sup_W[s][h]  (Hadamard factored out) ----
__global__ void qnn_supvec(const float* __restrict__ supW,
                           const float* __restrict__ coeffs,
                           float* __restrict__ supv) {
  int h = threadIdx.x;  // 512 threads
  float s = 0.f;
#pragma unroll
  for (int i = 0; i < 8; ++i) s += coeffs[i] * supW[i * HD + h];
  supv[h] = s;
}

// ---- per-quadruple 4x4 linear map: CNOT,strength,H,diag(sup), then 3 layers
//      of {H, I+px*PX, I+pz*PZ, CNOT}. Row-vector convention: y = v @ M. ----
__global__ void qnn_quadmap(const float* __restrict__ ent_strength,
                            const float* __restrict__ supv,
                            const float* __restrict__ cp,  // (3,8,3)
                            float* __restrict__ Mq) {      // (128,4,4)
  int g = threadIdx.x;  // 128 quadruples
  if (g >= 128) return;
  const float is2 = 0.70710678118654752440f;
  const int p = g >> 5;  // pair: 32 quads per 128-wide pair
  const int q = g >> 4;  // qubit: rep=64 elems -> 16 quads per qubit
  const float es = ent_strength[p];
  float sv0 = supv[4 * g + 0], sv1 = supv[4 * g + 1];
  float sv2 = supv[4 * g + 2], sv3 = supv[4 * g + 3];
#pragma unroll
  for (int i = 0; i < 4; ++i) {
    float v[4] = {0.f, 0.f, 0.f, 0.f};
    v[i] = 1.f;
    // entanglement tail: CNOT permute (swap 2,3) then strength scale
    float t = v[2]; v[2] = v[3]; v[3] = t;
    v[0] *= es; v[1] *= es; v[2] *= es; v[3] *= es;
    // superposition: Hadamard pairs, then elementwise sup_vec
    { float a = v[0], b = v[1]; v[0] = (a + b) * is2; v[1] = (a - b) * is2;
      a = v[2]; b = v[3];       v[2] = (a + b) * is2; v[3] = (a - b) * is2; }
    v[0] *= sv0; v[1] *= sv1; v[2] *= sv2; v[3] *= sv3;
    // variational layers (PY branch is exactly zero in the reference)
#pragma unroll
    for (int l = 0; l < 3; ++l) {
      const float px = cp[(l * 8 + q) * 3 + 0];
      const float pz = cp[(l * 8 + q) * 3 + 2];
      { float a = v[0], b = v[1]; v[0] = (a + b) * is2; v[1] = (a - b) * is2;
        a = v[2]; b = v[3];       v[2] = (a + b) * is2; v[3] = (a - b) * is2; }
      { float a = v[0], b = v[1]; v[0] = a + px * b; v[1] = b + px * a;
        a = v[2]; b = v[3];       v[2] = a + px * b; v[3] = b + px * a; }
      v[0] *= (1.f + pz); v[1] *= (1.f - pz);
      v[2] *= (1.f + pz); v[3] *= (1.f - pz);
      t = v[2]; v[2] = v[3]; v[3] = t;
    }
#pragma unroll
    for (int j = 0; j < 4; ++j) Mq[g * 16 + i * 4 + j] = v[j];
  }
}

// ---- T2 = M_quad_blockdiag @ T1  (4 MACs per output element) ----
__global__ void qnn_quadapply(const float* __restrict__ Mq,
                              const float* __restrict__ T1,
                              float* __restrict__ T2) {
  int idx = blockIdx.x * blockDim.x + threadIdx.x;  // h*512 + o
  if (idx >= HD * HD) return;
  const int o = idx & (HD - 1);
  const int h = idx >> 9;
  const int g = h >> 2, a = h & 3;
  const float* m = Mq + g * 16 + a * 4;
  const float* t = T1 + (size_t)(g * 4) * HD + o;
  T2[idx] = m[0] * t[0] + m[1] * t[HD] + m[2] * t[2 * HD] + m[3] * t[3 * HD];
}

// ---- b_eff[o] = in_b @ T3 + out_b ----
__global__ void qnn_bias(const float* __restrict__ in_b,
                         const float* __restrict__ T3,
                         const float* __restrict__ out_b,
                         float* __restrict__ beff) {
  int o = threadIdx.x;  // 512 threads
  float s = out_b[o];
  for (int h = 0; h < HD; ++h) s += in_b[h] * T3[(size_t)h * HD + o];
  beff[o] = s;
}

// ---------------------------------------------------------------------------
// fp32 GEMM: C[M,N] = A[M,K] @ B[K,N] (+ bias[N]), row-major, using
// V_WMMA_F32_16X16X4_F32. Block = 128 threads (4 waves, 2x2); block tile =
// 128x64; wave tile = 64x32 (8 v8f accumulators -> 8 WMMA : 8 loads per
// K-step). LDA/LDB/LDC are compile-time so every inner-loop load is a
// global_load with an immediate offset. M % 128 == 0, N % 64 == 0, K % 4 == 0.
// Fragment layouts per CDNA5 ISA 7.12.2:
//   A 16x4 : lanes 0-15 -> M, VGPR{0,1} = K{0,1} (lanes 0-15) / K{2,3} (16-31)
//   B 4x16 : lanes%16 -> N, VGPR{0,1} = K{0,1} (lanes 0-15) / K{2,3} (16-31)
//   C 16x16: lanes%16 -> N, VGPR r -> M = r (lanes 0-15) / r+8 (lanes 16-31)
// ---------------------------------------------------------------------------
template <int LDA, int LDB, int LDC>
__global__ __launch_bounds__(128) void qnn_gemm_wmma(
    const float* __restrict__ A, const float* __restrict__ B,
    const float* __restrict__ bias, float* __restrict__ C, int K) {
  const int tid  = threadIdx.x;
  const int lane = tid & 31;
  const int l16  = lane & 15;
  const int half = lane >> 4;
  const int wid  = tid >> 5;
  const int m0 = blockIdx.y * 128 + (wid & 1) * 64;
  const int n0 = blockIdx.x * 64 + (wid >> 1) * 32;

  const v8f vzero = {0.f, 0.f, 0.f, 0.f, 0.f, 0.f, 0.f, 0.f};
  v8f acc[4][2];
#pragma unroll
  for (int mi = 0; mi < 4; ++mi) { acc[mi][0] = vzero; acc[mi][1] = vzero; }

  // per-lane base pointers (computed once; inner loop uses immediate offsets)
  const float* Ap[4];
#pragma unroll
  for (int mi = 0; mi < 4; ++mi)
    Ap[mi] = A + (size_t)(m0 + mi * 16 + l16) * LDA + 2 * half;
  const float* B0 = B + (size_t)(2 * half) * LDB + n0 + l16;
  const float* B1 = B0 + 16;

#pragma unroll 4
  for (int k = 0; k < K; k += 4) {
    v2f a[4];
#pragma unroll
    for (int mi = 0; mi < 4; ++mi)
      a[mi] = *reinterpret_cast<const v2f*>(Ap[mi] + k);
    v2f b0, b1;
    b0.x = B0[(size_t)k * LDB];
    b0.y = B0[(size_t)(k + 1) * LDB];
    b1.x = B1[(size_t)k * LDB];
    b1.y = B1[(size_t)(k + 1) * LDB];
#pragma unroll
    for (int mi = 0; mi < 4; ++mi) {
      acc[mi][0] = __builtin_amdgcn_wmma_f32_16x16x4_f32(
          false, a[mi], false, b0, (short)0, acc[mi][0], false, false);
      acc[mi][1] = __builtin_amdgcn_wmma_f32_16x16x4_f32(
          false, a[mi], false, b1, (short)0, acc[mi][1], false, false);
    }
  }

#pragma unroll
  for (int mi = 0; mi < 4; ++mi)
#pragma unroll
    for (int ni = 0; ni < 2; ++ni) {
      const int col = n0 + ni * 16 + l16;
      const float bv = bias ? bias[col] : 0.f;
      float* Cp = C + (size_t)(m0 + mi * 16 + half * 8) * LDC + col;
#pragma unroll
      for (int r = 0; r < 8; ++r) Cp[(size_t)r * LDC] = acc[mi][ni][r] + bv;
    }
}

// ---------------------------------------------------------------------------
extern "C" void kernel_launch(void* const* d_in, const int* in_sizes, int n_in,
                              void* d_out, int out_size, void* d_ws, size_t ws_size,
                              hipStream_t stream) {
  const float* x       = (const float*)d_in[0];   // (8,2048,512)
  const float* in_W    = (const float*)d_in[1];   // (512,512)
  const float* in_b    = (const float*)d_in[2];   // (512,)
  const float* ent_W   = (const float*)d_in[3];   // (4,128,128)
  const float* ent_str = (const float*)d_in[4];   // (4,)
  const float* sup_W   = (const float*)d_in[5];   // (8,512)
  const float* sup_c   = (const float*)d_in[6];   // (8,)
  const float* cparams = (const float*)d_in[7];   // (3,8,3)
  const float* meas    = (const float*)d_in[8];   // (512,512)
  const float* out_W   = (const float*)d_in[9];   // (512,512)
  const float* out_b   = (const float*)d_in[10];  // (512,)
  float* out = (float*)d_out;                     // (16384,512)

  // workspace layout (floats): 3 * 512*512 matrices + small vectors (~3.2 MB)
  float* ws   = (float*)d_ws;
  float* supv = ws;                 // 512
  float* Mq   = ws + 512;           // 128*16
  float* beff = ws + 2560;          // 512
  float* bufA = ws + 3072;          // 512*512  T1, later reused for W_eff
  float* bufB = bufA + HD * HD;     // 512*512  T2
  float* bufC = bufB + HD * HD;     // 512*512  T3

  // 1) sup_vec and the 128 per-quadruple 4x4 maps
  qnn_supvec<<<1, 512, 0, stream>>>(sup_W, sup_c, supv);
  qnn_quadmap<<<1, 128, 0, stream>>>(ent_str, supv, cparams, Mq);

  // 2) T1 = meas @ out_W                      (512x512x512, WMMA)
  qnn_gemm_wmma<HD, HD, HD><<<dim3(8, 4), 128, 0, stream>>>(
      meas, out_W, nullptr, bufA, HD);
  // 3) T2 = M_quad @ T1                       (block-diag 4x4, cheap)
  qnn_quadapply<<<(HD * HD) / 256, 256, 0, stream>>>(Mq, bufA, bufB);
  // 4) T3 = E_blockdiag @ T2                  (4 x [128x128x512], WMMA)
  for (int p = 0; p < 4; ++p)
    qnn_gemm_wmma<128, HD, HD><<<dim3(8, 1), 128, 0, stream>>>(
        ent_W + (size_t)p * 128 * 128, bufB + (size_t)p * 128 * HD, nullptr,
        bufC + (size_t)p * 128 * HD, 128);
  // 5) W_eff = in_W @ T3                      (512x512x512, WMMA)
  qnn_gemm_wmma<HD, HD, HD><<<dim3(8, 4), 128, 0, stream>>>(
      in_W, bufC, nullptr, bufA, HD);
  // 6) b_eff = in_b @ T3 + out_b
  qnn_bias<<<1, 512, 0, stream>>>(in_b, bufC, out_b, beff);

  // 7) main compute-bound GEMM: out = x @ W_eff + b_eff  (16384x512x512)
  qnn_gemm_wmma<HD, HD, HD><<<dim3(HD / 64, NROW / 128), 128, 0, stream>>>(
      x, bufA, beff, out, HD);
}